// PixelLinkLoss_14800457302079
// MI455X (gfx1250) — compile-verified
//
#include <hip/hip_runtime.h>
#include <hip/hip_bf16.h>
#include <math.h>

#define BATCH 16
#define HW    65536            // 256*256
#define CH    10
#define BLK   256
#define NPIX  (BATCH * HW)

typedef __attribute__((ext_vector_type(2))) float v2f;
typedef __attribute__((ext_vector_type(8))) float v8f;

// Exact f32 wave32 sum using the matrix pipe:
// A(16x4): lane m vgpr0 = x[m] (lanes0-15) / x[m+16] (lanes16-31), vgpr1 = 0.
// B = ones  =>  D[m,n] = x[m] + x[m+16] for every n.
// D vgpr r: lanes0-15 hold row r, lanes16-31 hold row r+8; summing the 8
// D regs gives half-wave sums; one ds_swizzle SWAPX16 completes the 32-lane sum.
__device__ __forceinline__ float wave_sum32(float x) {
  v2f a; a.x = x;    a.y = 0.0f;
  v2f b; b.x = 1.0f; b.y = 1.0f;
  v8f c = {};
  v8f d = __builtin_amdgcn_wmma_f32_16x16x4_f32(false, a, false, b,
                                                (short)0, c, false, false);
  float s = d[0] + d[1] + d[2] + d[3] + d[4] + d[5] + d[6] + d[7];
  // lane ^= 16 exchange: group-of-32 swizzle, and=0x1F or=0 xor=0x10
  int sw = __builtin_amdgcn_ds_swizzle(__float_as_int(s), 0x401F);
  return s + __int_as_float(sw);
}

__global__ void init_ws(unsigned* zbase, int nwords) {
  int i = blockIdx.x * blockDim.x + threadIdx.x;
  if (i < nwords) zbase[i] = 0u;
}

// Main fused pass: stages target rows into LDS with async-to-LDS copies,
// computes cls CE + neg score + 4 link CEs, builds level-0 radix histogram,
// and block-reduces 7 scalar accumulators via WMMA wave sums.
__global__ void __launch_bounds__(BLK)
main_pass(const float* __restrict__ target, const float* __restrict__ logits,
          float* __restrict__ pneg, float* __restrict__ clsce,
          unsigned* __restrict__ histL0, float* __restrict__ cnt,
          float* __restrict__ g)
{
  __shared__ float ds_t[BLK * CH];   // 2560 floats = 10240 B
  __shared__ float lacc[8];
  const int tid = threadIdx.x;
  const int blk = blockIdx.x;
  const int b   = blk >> 8;                 // 256 blocks per image
  const size_t pix0 = (size_t)blk * BLK;

  // Async stage of this block's 2560 contiguous target floats (640 x 16B).
  {
    const float* gbase = target + pix0 * CH;
    unsigned lds0 = (unsigned)(size_t)(void*)ds_t;   // low 32 bits = LDS offset
    for (int cidx = tid; cidx < (BLK * CH) / 4; cidx += BLK) {
      const float* gp = gbase + (size_t)cidx * 4;
      unsigned lo = lds0 + (unsigned)cidx * 16u;
      asm volatile("global_load_async_to_lds_b128 %0, %1, off"
                   :: "v"(lo), "v"(gp) : "memory");
    }
  }
  if (tid < 8) lacc[tid] = 0.0f;
  asm volatile("s_wait_asynccnt 0x0" ::: "memory");
  __syncthreads();

  const size_t p = pix0 + tid;
  const int n = (int)(p & (HW - 1));
  const float* t = &ds_t[tid * CH];
  const float* lbase = logits + (size_t)b * CH * HW + n;

  float lab = t[0], wgt = t[1];
  bool pos = lab > 0.0f;
  bool neg = lab == 0.0f;

  // 2-class log-softmax for cls head
  float l0 = lbase[0];
  float l1 = lbase[(size_t)HW];
  float mx = fmaxf(l0, l1);
  float e0 = __expf(l0 - mx), e1 = __expf(l1 - mx);
  float z  = e0 + e1;
  float lse = mx + __logf(z);
  float ce  = -((pos ? l1 : l0) - lse);
  float score = e0 / z;                                // softmax class-0 prob
  float pn = neg ? score : __int_as_float(0x7f800000); // +inf marks non-neg

  pneg[p]  = pn;
  clsce[p] = ce;
  atomicAdd(&histL0[(b << 11) + (__float_as_uint(pn) >> 21)], 1u);

  float wce = ce * wgt;
  float s1 = 0.f, c1v = 0.f, s0 = 0.f, c0v = 0.f;
#pragma unroll
  for (int j = 0; j < 4; ++j) {
    float la = lbase[(size_t)(2 + j) * HW];          // link class 0
    float lb = lbase[(size_t)(6 + j) * HW];          // link class 1
    float m2 = fmaxf(la, lb);
    float lse2 = m2 + __logf(__expf(la - m2) + __expf(lb - m2));
    float lbl = t[2 + j];
    float lwt = t[6 + j];
    float one  = (lbl > 0.5f) ? 1.0f : 0.0f;         // branchless masks
    float zero = 1.0f - one;
    float lce  = -((lbl > 0.5f ? lb : la) - lse2);
    float wl   = lwt * lce;
    s1  = fmaf(wl,  one,  s1);
    c1v = fmaf(lwt, one,  c1v);
    s0  = fmaf(wl,  zero, s0);
    c0v = fmaf(lwt, zero, c0v);
  }

  float q[7] = { wce, s1, c1v, s0, c0v, pos ? 1.0f : 0.0f, neg ? 1.0f : 0.0f };
#pragma unroll
  for (int i = 0; i < 7; ++i) {
    float ws = wave_sum32(q[i]);                     // EXEC all-ones here
    if ((tid & 31) == 0) atomicAdd(&lacc[i], ws);
  }
  __syncthreads();
  if (tid < 5)       atomicAdd(&g[tid], lacc[tid]);
  else if (tid == 5) atomicAdd(&cnt[2 * b + 0], lacc[5]);
  else if (tid == 6) atomicAdd(&cnt[2 * b + 1], lacc[6]);
}

// Per-image k = min(3*n_pos, n_neg); use_all = (n_pos==0 || k==0).
__global__ void setup_select(const float* __restrict__ cnt,
                             unsigned* __restrict__ state) {
  int b = threadIdx.x;
  if (b >= BATCH) return;
  unsigned npos = (unsigned)(cnt[2 * b + 0] + 0.5f);
  unsigned nneg = (unsigned)(cnt[2 * b + 1] + 0.5f);
  unsigned k = 3u * npos; if (nneg < k) k = nneg;
  unsigned use_all = (npos == 0u || k == 0u) ? 1u : 0u;
  unsigned* st = state + b * 4;
  st[0] = 0u;       // prefix (high bits found so far)
  st[1] = k;        // remaining rank within current prefix
  st[2] = use_all;
  st[3] = 0u;       // final threshold bits
}

// Per-image serial scan of one radix level's histogram (16 tiny scans).
__global__ void scan_level(const unsigned* __restrict__ hist, unsigned bins,
                           unsigned width, unsigned* __restrict__ state,
                           int isLast) {
  int b = threadIdx.x;
  if (b >= BATCH) return;
  unsigned* st = state + b * 4;
  if (st[2]) return;
  unsigned k = st[1];
  const unsigned* h = hist + (size_t)b * bins;
  unsigned cum = 0, d = 0;
  for (unsigned i = 0; i < bins; ++i) {
    unsigned cbin = h[i];
    if (cum + cbin >= k) { d = i; break; }
    cum += cbin;
  }
  st[0] = (st[0] << width) | d;
  st[1] = k - cum;
  if (isLast) st[3] = st[0];
}

// Filtered histogram for refinement levels (reads cached scores only: 4 MB).
__global__ void __launch_bounds__(BLK)
hist_filter(const float* __restrict__ pneg, const unsigned* __restrict__ state,
            unsigned* __restrict__ hist, int level)
{
  size_t p = (size_t)blockIdx.x * BLK + threadIdx.x;
  int b = (int)(p >> 16);
  const unsigned* st = state + b * 4;
  if (st[2]) return;
  unsigned bits = __float_as_uint(pneg[p]);
  if (level == 1) {
    if ((bits >> 21) == st[0])
      atomicAdd(&hist[(b << 11) + ((bits >> 10) & 0x7FFu)], 1u);
  } else {
    if ((bits >> 10) == st[0])
      atomicAdd(&hist[(b << 10) + (bits & 0x3FFu)], 1u);
  }
}

// Apply OHEM selection, accumulate selected CE and count.
__global__ void __launch_bounds__(BLK)
final_pass(const float* __restrict__ pneg, const float* __restrict__ clsce,
           const unsigned* __restrict__ state, float* __restrict__ g)
{
  __shared__ float lacc[2];
  int tid = threadIdx.x;
  size_t p = (size_t)blockIdx.x * BLK + tid;
  int b = (int)(p >> 16);
  if (tid < 2) lacc[tid] = 0.f;
  __syncthreads();
  const unsigned* st = state + b * 4;
  unsigned bits = __float_as_uint(pneg[p]);
  bool is_neg = (bits != 0x7f800000u);
  bool sel = st[2] ? is_neg : (is_neg && bits <= st[3]); // uint order == float order (>=0)
  float ws = wave_sum32(sel ? clsce[p] : 0.f);
  if ((tid & 31) == 0) atomicAdd(&lacc[0], ws);
  ws = wave_sum32(sel ? 1.0f : 0.0f);
  if ((tid & 31) == 0) atomicAdd(&lacc[1], ws);
  __syncthreads();
  if (tid == 0) atomicAdd(&g[5], lacc[0]);
  if (tid == 1) atomicAdd(&g[6], lacc[1]);
}

__global__ void finalize(const float* __restrict__ cnt,
                         const float* __restrict__ g, float* __restrict__ out) {
  float npos = 0.f;
  for (int b = 0; b < BATCH; ++b) npos += cnt[2 * b];
  float cls = (g[0] + g[5]) / (npos + g[6]);
  out[0] = 2.0f * cls;
  float link = g[1] / g[2] + g[3] / g[4];
  out[1] = (npos == 0.f) ? 0.f : link;
}

extern "C" void kernel_launch(void* const* d_in, const int* in_sizes, int n_in,
                              void* d_out, int out_size, void* d_ws, size_t ws_size,
                              hipStream_t stream) {
  (void)in_sizes; (void)n_in; (void)out_size; (void)ws_size;
  const float* target = (const float*)d_in[0];
  const float* logits = (const float*)d_in[1];
  float* out = (float*)d_out;

  char* ws = (char*)d_ws;
  float*    pneg   = (float*)(ws);                         // 4 MB
  float*    clsce  = (float*)(ws + 4ull * NPIX);           // 4 MB
  unsigned* histL0 = (unsigned*)(ws + 8ull * NPIX);        // 16*2048
  unsigned* histL1 = histL0 + BATCH * 2048;                // 16*2048
  unsigned* histL2 = histL1 + BATCH * 2048;                // 16*1024
  float*    cnt    = (float*)(histL2 + BATCH * 1024);      // 16*2
  float*    g      = cnt + BATCH * 2;                      // 8
  unsigned* state  = (unsigned*)(g + 8);                   // 16*4

  int zwords = BATCH * 2048 * 2 + BATCH * 1024 + BATCH * 2 + 8 + BATCH * 4;
  init_ws<<<(zwords + 255) / 256, 256, 0, stream>>>(histL0, zwords);

  main_pass<<<NPIX / BLK, BLK, 0, stream>>>(target, logits, pneg, clsce,
                                            histL0, cnt, g);
  setup_select<<<1, 32, 0, stream>>>(cnt, state);
  scan_level<<<1, 32, 0, stream>>>(histL0, 2048, 11, state, 0);
  hist_filter<<<NPIX / BLK, BLK, 0, stream>>>(pneg, state, histL1, 1);
  scan_level<<<1, 32, 0, stream>>>(histL1, 2048, 11, state, 0);
  hist_filter<<<NPIX / BLK, BLK, 0, stream>>>(pneg, state, histL2, 2);
  scan_level<<<1, 32, 0, stream>>>(histL2, 1024, 10, state, 1);
  final_pass<<<NPIX / BLK, BLK, 0, stream>>>(pneg, clsce, state, g);
  finalize<<<1, 1, 0, stream>>>(cnt, g, out);
}